// Attention_19877108646606
// MI455X (gfx1250) — compile-verified
//
#include <hip/hip_runtime.h>
#include <hip/hip_bf16.h>
#include <math.h>

// Problem constants (B,T,H) from the reference.
#define BB 128
#define TT 1024
#define HH 512

#define TCHUNK 128             // t rows handled per (energy/context) block
#define NCHUNK (TT / TCHUNK)   // 8 t-chunks

// --- CDNA5 WMMA f32 16x16x4 (probe-verified in round 1: lowers to
// v_wmma_f32_16x16x4_f32) --------------------------------------------------
typedef __attribute__((ext_vector_type(2))) float v2f;
typedef __attribute__((ext_vector_type(8))) float v8f;

#if defined(__gfx1250__) && __has_builtin(__builtin_amdgcn_wmma_f32_16x16x4_f32)
#define HAVE_WMMA_F32 1
#endif

__device__ __forceinline__ v8f wmma_f32_16x16x4(v2f a, v2f b, v8f c) {
#ifdef HAVE_WMMA_F32
  return __builtin_amdgcn_wmma_f32_16x16x4_f32(false, a, false, b, (short)0, c,
                                               false, false);
#else
  // Compile-safety fallback only (not layout-equivalent).
  c[0] += a.x * b.x + a.y * b.y;
  return c;
#endif
}

// --- CDNA5 async global->LDS (GLOBAL_LOAD_ASYNC_TO_LDS_B128, ASYNCcnt) -----
// clang-22 declares the builtin as taking <4 x i32> pointers:
//   arg0: v4i addrspace(1)*  (printed as "int __device__ *" in HIP mode)
//   arg1: v4i addrspace(3)*  (LDS destination)
//   arg2/arg3: i32 offset / cache-policy immediates
#if defined(__gfx1250__) &&                                          \
    __has_builtin(__builtin_amdgcn_global_load_async_to_lds_b128) && \
    __has_builtin(__builtin_amdgcn_s_wait_asynccnt)
#define HAVE_ASYNC_LDS 1
#endif

typedef int v4i_ __attribute__((vector_size(16)));

#ifdef HAVE_ASYNC_LDS
#define ASYNC_COPY16(src, dst)                                        \
  __builtin_amdgcn_global_load_async_to_lds_b128(                     \
      (__attribute__((address_space(1))) v4i_*)(src),                 \
      (__attribute__((address_space(3))) v4i_*)(dst), 0, 0)
#define ASYNC_WAIT(n) __builtin_amdgcn_s_wait_asynccnt(n)
#else
#define ASYNC_COPY16(src, dst) (*(float4*)(dst) = *(const float4*)(src))
#define ASYNC_WAIT(n) ((void)0)
#endif

// ============================================================================
// K1: energy[b,t] = <query[b,:], key[t,b,:]>
// One block per (t-chunk, b). Key tiles (128t x 64h) streamed into
// double-buffered LDS with async global->LDS; each wave owns 16 t's.
// WMMA A = q chunk broadcast to all 16 rows, B = key columns.
// D row M=0 (c[0], lanes 0..15) == the wave's 16 energies.
// ============================================================================
#define K1_HT 64
#define K1_STRIDE (K1_HT + 4)   // 68 floats: lanes 0..15 hit banks 4n
#define K1_INFLIGHT 8           // async-load instructions issued per wave/tile

__global__ __launch_bounds__(256) void energy_kernel(
    const float* __restrict__ q, const float* __restrict__ key,
    float* __restrict__ energy) {
  __shared__ float qs[HH];
  __shared__ float ks[2][TCHUNK * K1_STRIDE];   // 2 x 34 KB

  const int b    = blockIdx.y;
  const int t0   = blockIdx.x * TCHUNK;
  const int tid  = threadIdx.x;
  const int lane = tid & 31;
  const int wave = tid >> 5;            // 0..7, wave32

  // Stage the query row once (2 KB, reused by all 8 waves).
  for (int i = tid; i < HH; i += 256) qs[i] = q[(size_t)b * HH + i];

  const int hq = (tid & 15) << 2;       // 16 threads x 16B cover a 64-float row
  const int r0 = tid >> 4;              // 16 rows per sweep, 8 rows per thread

  const int n  = lane & 15;             // N (t within wave tile) for B/C/D
  const int kb = (lane >> 4) << 1;      // lanes 0..15 hold K=0,1; 16..31 K=2,3
  const int trow = wave * 16 + n;

#define K1_ISSUE(buf, h0c)                                                  \
  do {                                                                      \
    const size_t base_ = (size_t)b * HH + (h0c) + hq;                       \
    for (int r = r0; r < TCHUNK; r += 16)                                   \
      ASYNC_COPY16(&key[(size_t)(t0 + r) * (BB * HH) + base_],              \
                   &ks[buf][r * K1_STRIDE + hq]);                           \
  } while (0)

  K1_ISSUE(0, 0);                       // prime the pipeline
  int cur = 0;
  v8f c = {};
  for (int h0 = 0; h0 < HH; h0 += K1_HT) {
    if (h0 + K1_HT < HH) {
      K1_ISSUE(cur ^ 1, h0 + K1_HT);    // stream next tile behind compute
      ASYNC_WAIT(K1_INFLIGHT);          // wait only for the *current* tile
    } else {
      ASYNC_WAIT(0);
    }
    __syncthreads();                    // current tile visible to all waves

    const float* kt_ = ks[cur];
    for (int h = 0; h < K1_HT; h += 4) {
      v2f a;  a.x = qs[h0 + h + kb];  a.y = qs[h0 + h + kb + 1];
      v2f bv = *(const v2f*)&kt_[trow * K1_STRIDE + h + kb];
      c = wmma_f32_16x16x4(a, bv, c);
    }
    __syncthreads();                    // tile consumed; safe to overwrite
    cur ^= 1;
  }
#undef K1_ISSUE

  if (lane < 16)   // D VGPR0, lanes 0..15 = row M=0
    energy[(size_t)b * TT + t0 + wave * 16 + lane] = c[0];
}

// ============================================================================
// K2: softmax over t, mask, renormalize. One block per b. Deterministic
// tree reductions in LDS (no wave-width assumptions, no atomics).
// ============================================================================
__global__ __launch_bounds__(256) void softmax_kernel(
    const float* __restrict__ energy, const int* __restrict__ mask,
    float* __restrict__ attn) {
  __shared__ float row[TT];
  __shared__ float sred[256];
  const int b = blockIdx.x;
  const int tid = threadIdx.x;

  float lmax = -INFINITY;
  for (int t = tid; t < TT; t += 256) {
    const float e = energy[(size_t)b * TT + t];
    row[t] = e;
    lmax = fmaxf(lmax, e);
  }
  sred[tid] = lmax; __syncthreads();
  for (int s = 128; s > 0; s >>= 1) {
    if (tid < s) sred[tid] = fmaxf(sred[tid], sred[tid + s]);
    __syncthreads();
  }
  const float gmax = sred[0]; __syncthreads();

  float lsum = 0.f;
  for (int t = tid; t < TT; t += 256) {
    const float e = __expf(row[t] - gmax);
    row[t] = e;
    lsum += e;
  }
  sred[tid] = lsum; __syncthreads();
  for (int s = 128; s > 0; s >>= 1) {
    if (tid < s) sred[tid] += sred[tid + s];
    __syncthreads();
  }
  const float gsum = sred[0]; __syncthreads();

  float lms = 0.f;
  for (int t = tid; t < TT; t += 256) {
    const float w = (float)mask[(size_t)b * TT + t] * (row[t] / gsum);
    row[t] = w;
    lms += w;
  }
  sred[tid] = lms; __syncthreads();
  for (int s = 128; s > 0; s >>= 1) {
    if (tid < s) sred[tid] += sred[tid + s];
    __syncthreads();
  }
  const float gms = sred[0]; __syncthreads();

  const float inv = 1.f / gms;
  for (int t = tid; t < TT; t += 256)
    attn[(size_t)b * TT + t] = row[t] * inv;
}

// ============================================================================
// K3: context partials. One block per (t-chunk, b). Value streamed as
// 64t x 128h sub-tiles (flat pipelined tile sequence, double-buffered,
// async global->LDS). Each wave owns a 16-h column stripe; WMMA A = weight
// chunk broadcast, B = value columns. Accumulator is carried across the two
// t-sub-tiles of each h-stripe, then flushed to per-chunk partials
// (fixed-order summation -> deterministic, no float atomics).
// ============================================================================
#define K3_HT 128                      // h columns per stripe (8 waves x 16)
#define K3_TSUB 64                     // t rows per LDS sub-tile
#define K3_STRIDE (K3_HT + 4)          // 132 floats
#define K3_INFLIGHT 8                  // async-load instructions per wave/tile
#define K3_NTILES ((HH / K3_HT) * (TCHUNK / K3_TSUB))   // 4 * 2 = 8

__global__ __launch_bounds__(256) void context_kernel(
    const float* __restrict__ val, const float* __restrict__ attn,
    float* __restrict__ partial) {
  __shared__ float wsh[TCHUNK];
  __shared__ float vs[2][K3_TSUB * K3_STRIDE];   // 2 x 33.8 KB

  const int b    = blockIdx.y;
  const int t0   = blockIdx.x * TCHUNK;
  const int tid  = threadIdx.x;
  const int lane = tid & 31;
  const int wave = tid >> 5;

  for (int i = tid; i < TCHUNK; i += 256)
    wsh[i] = attn[(size_t)b * TT + t0 + i];

  const int hq = (tid & 31) << 2;      // 32 threads x 16B cover a 128-float row
  const int r0 = tid >> 5;             // 8 rows per sweep, 8 rows per thread

  const int n  = lane & 15;
  const int kb = (lane >> 4) << 1;
  const int hcol = wave * 16 + n;

  // Tile it: h0 = (it>>1)*128, tt = (it&1)*64
#define K3_ISSUE(buf, it)                                                    \
  do {                                                                       \
    const int h0_ = ((it) >> 1) * K3_HT;                                     \
    const int tt_ = ((it) & 1) * K3_TSUB;                                    \
    const size_t base_ = (size_t)b * HH + h0_ + hq;                          \
    for (int r = r0; r < K3_TSUB; r += 8)                                    \
      ASYNC_COPY16(&val[(size_t)(t0 + tt_ + r) * (BB * HH) + base_],         \
                   &vs[buf][r * K3_STRIDE + hq]);                            \
  } while (0)

  K3_ISSUE(0, 0);                      // prime the pipeline
  int cur = 0;
  v8f c = {};
  for (int it = 0; it < K3_NTILES; ++it) {
    if (it + 1 < K3_NTILES) {
      K3_ISSUE(cur ^ 1, it + 1);
      ASYNC_WAIT(K3_INFLIGHT);
    } else {
      ASYNC_WAIT(0);
    }
    __syncthreads();

    const int tt = (it & 1) * K3_TSUB;
    const float* vt_ = vs[cur];
    for (int kt = 0; kt < K3_TSUB; kt += 4) {
      v2f a;  a.x = wsh[tt + kt + kb];  a.y = wsh[tt + kt + kb + 1];
      v2f bv; bv.x = vt_[(kt + kb) * K3_STRIDE + hcol];
              bv.y = vt_[(kt + kb + 1) * K3_STRIDE + hcol];
      c = wmma_f32_16x16x4(a, bv, c);
    }
    __syncthreads();

    if (it & 1) {                      // finished both t-sub-tiles of stripe
      const int h0 = (it >> 1) * K3_HT;
      if (lane < 16)
        partial[((size_t)b * NCHUNK + blockIdx.x) * HH + h0 + wave * 16 +
                lane] = c[0];
      c = (v8f){};
    }
    cur ^= 1;
  }
#undef K3_ISSUE
}

// ============================================================================
// K4: context[b,h] = sum over the 8 t-chunk partials (fixed order).
// ============================================================================
__global__ __launch_bounds__(256) void reduce_kernel(
    const float* __restrict__ partial, float* __restrict__ ctx) {
  const int i = blockIdx.x * 256 + threadIdx.x;
  if (i >= BB * HH) return;
  const int b = i / HH, h = i - b * HH;
  float s = 0.f;
  for (int c = 0; c < NCHUNK; ++c)
    s += partial[((size_t)b * NCHUNK + c) * HH + h];
  ctx[i] = s;
}

// ============================================================================
extern "C" void kernel_launch(void* const* d_in, const int* in_sizes, int n_in,
                              void* d_out, int out_size, void* d_ws,
                              size_t ws_size, hipStream_t stream) {
  const float* q    = (const float*)d_in[0];   // (B,H)
  const float* key  = (const float*)d_in[1];   // (T,B,H)
  const float* val  = (const float*)d_in[2];   // (T,B,H)
  const int*   mask = (const int*)d_in[3];     // (B,1,T)

  float* out  = (float*)d_out;
  float* ctx  = out;               // context: B*H floats (first tuple output)
  float* attn = out + BB * HH;     // masked_attention: B*T floats (second)

  float* energy  = (float*)d_ws;             // B*T   = 512 KB
  float* partial = energy + (size_t)BB * TT; // B*8*H = 2 MB

  energy_kernel<<<dim3(NCHUNK, BB), 256, 0, stream>>>(q, key, energy);
  softmax_kernel<<<dim3(BB), 256, 0, stream>>>(energy, mask, attn);
  context_kernel<<<dim3(NCHUNK, BB), 256, 0, stream>>>(val, attn, partial);
  reduce_kernel<<<dim3((BB * HH + 255) / 256), 256, 0, stream>>>(partial, ctx);
}